// MeanAggregator_70970039599213
// MI455X (gfx1250) — compile-verified
//
#include <hip/hip_runtime.h>
#include <hip/hip_bf16.h>

// MeanAggregator: out[i,:] = (1/32) * sum_k features[neigh_idx[i,k], :]
// BATCH=50000, K=32, D=128, fp32.
//
// Bandwidth-bound gather (819 MB logical gather, table L2-resident on MI455X's
// 192 MB L2). Mapping: one wave32 per target node, lane l owns the 16-byte
// slice [4l, 4l+4) of the 128-float row, so every neighbor row is one
// perfectly coalesced wave-wide global_load_b128 (512 B).

#define KNBR 32
#define DIM  128
#define WAVES_PER_BLOCK 8
#define TPB (WAVES_PER_BLOCK * 32)

typedef float v4f __attribute__((ext_vector_type(4)));
typedef int   v4i __attribute__((ext_vector_type(4)));

#if defined(__has_builtin)
#if __has_builtin(__builtin_amdgcn_global_load_async_to_lds_b128) && \
    __has_builtin(__builtin_amdgcn_s_wait_asynccnt)
#define USE_ASYNC_LDS 1
// Builtin signature (per hipcc diagnostic): non-const v4i pointers,
// (global src AS1, lds dst AS3, imm offset, imm cpol).
typedef __attribute__((address_space(1))) v4i gv4i;
typedef __attribute__((address_space(3))) v4i lv4i;
#endif
#endif

__global__ __launch_bounds__(TPB) void MeanAggregator_kernel(
    const float* __restrict__ feat,   // [N_UNIQUE, 128]
    const int*   __restrict__ nidx,   // [BATCH, 32]
    float*       __restrict__ out,    // [BATCH, 128]
    int batch)
{
    __shared__ int sidx[WAVES_PER_BLOCK * KNBR];   // 1 KB index slab

    const int tid  = threadIdx.x;
    const int wave = tid >> 5;
    const int lane = tid & 31;
    const long long node0 = (long long)blockIdx.x * WAVES_PER_BLOCK;

    // ---- Stage this block's 8x32 int32 index slab into LDS (1 KB). ----
#if USE_ASYNC_LDS
    // 64 lanes x b128 = 1 KB, via the gfx1250 async global->LDS path
    // (tracked with ASYNCcnt, no VGPR return data).
    if (tid < (WAVES_PER_BLOCK * KNBR) / 4) {
        gv4i* gsrc = (gv4i*)(nidx + node0 * KNBR + tid * 4);
        lv4i* ldst = (lv4i*)(&sidx[tid * 4]);
        __builtin_amdgcn_global_load_async_to_lds_b128(gsrc, ldst, 0, 0);
    }
    __builtin_amdgcn_s_wait_asynccnt(0);   // issuing waves drain their async ops
    __syncthreads();                        // publish LDS to all 8 waves
#else
    if (tid < (WAVES_PER_BLOCK * KNBR) / 4) {
        ((int4*)sidx)[tid] = ((const int4*)(nidx + node0 * KNBR))[tid];
    }
    __syncthreads();
#endif

    const long long node = node0 + wave;
    if (node < batch) {
        // Lane l holds the index of neighbor l of this node.
        const int myidx = sidx[wave * KNBR + lane];

        // Prefetch this lane's entire neighbor row (4 x 128B cachelines):
        // across the wave that is the node's full 16 KB gather footprint,
        // issued before the serial accumulate loop (global_prefetch_b8,
        // no counters consumed).
        const float* myrow = feat + (size_t)(unsigned)myidx * DIM;
        __builtin_prefetch(myrow +  0);
        __builtin_prefetch(myrow + 32);
        __builtin_prefetch(myrow + 64);
        __builtin_prefetch(myrow + 96);

        v4f acc0 = {0.f, 0.f, 0.f, 0.f};
        v4f acc1 = acc0, acc2 = acc0, acc3 = acc0;

#pragma unroll
        for (int k = 0; k < KNBR; k += 4) {
            // v_readlane with immediate lane -> row index lands in an SGPR,
            // so the gather uses the SADDR form of global_load_b128
            // (scalar 64-bit base + per-lane 16B offset).
            const int j0 = __builtin_amdgcn_readlane(myidx, k + 0);
            const int j1 = __builtin_amdgcn_readlane(myidx, k + 1);
            const int j2 = __builtin_amdgcn_readlane(myidx, k + 2);
            const int j3 = __builtin_amdgcn_readlane(myidx, k + 3);

            const v4f v0 = ((const v4f*)(feat + (size_t)(unsigned)j0 * DIM))[lane];
            const v4f v1 = ((const v4f*)(feat + (size_t)(unsigned)j1 * DIM))[lane];
            const v4f v2 = ((const v4f*)(feat + (size_t)(unsigned)j2 * DIM))[lane];
            const v4f v3 = ((const v4f*)(feat + (size_t)(unsigned)j3 * DIM))[lane];

            acc0 += v0;
            acc1 += v1;
            acc2 += v2;
            acc3 += v3;
        }

        v4f r = (acc0 + acc1) + (acc2 + acc3);
        r *= (1.0f / (float)KNBR);

        // Output is written once and never re-read: non-temporal store keeps
        // L2 capacity for the (fully L2-resident) feature table.
        __builtin_nontemporal_store(r, (v4f*)(out + node * DIM) + lane);
    }
}

extern "C" void kernel_launch(void* const* d_in, const int* in_sizes, int n_in,
                              void* d_out, int out_size, void* d_ws, size_t ws_size,
                              hipStream_t stream) {
    const float* feat = (const float*)d_in[0];     // [100000, 128] fp32
    const int*   nidx = (const int*)d_in[1];       // [50000, 32] int
    float*       out  = (float*)d_out;             // [50000, 128] fp32

    const int batch  = in_sizes[1] / KNBR;         // 50000
    const int blocks = (batch + WAVES_PER_BLOCK - 1) / WAVES_PER_BLOCK;  // 6250

    MeanAggregator_kernel<<<blocks, TPB, 0, stream>>>(feat, nidx, out, batch);
}